// IGNN_74217034875084
// MI455X (gfx1250) — compile-verified
//
#include <hip/hip_runtime.h>

#define NN 50000
#define NE 800000
#define MT 3125           // NN / 16  (exact)
#define GEMM_BLOCKS 782   // ceil(3125 / 4 waves)
#define KAPPA_F 0.9f

typedef float v2f __attribute__((ext_vector_type(2)));
typedef float v8f __attribute__((ext_vector_type(8)));

// ---------------------------------------------------------------- utilities
__global__ void k_zero(float* __restrict__ p, int n) {
  int i = blockIdx.x * blockDim.x + threadIdx.x;
  if (i < n) p[i] = 0.0f;
}

__global__ void k_fill(float* __restrict__ p, int n, float c) {
  int i = blockIdx.x * blockDim.x + threadIdx.x;
  if (i < n) p[i] = c;
}

// features [128, NN] (feature-major) -> x [NN, 128] (node-major)
__global__ void k_transpose_feat(const float* __restrict__ F, float* __restrict__ X) {
  int i = blockIdx.x * blockDim.x + threadIdx.x;
  if (i < NN * 128) {
    int n = i >> 7, f = i & 127;
    X[i] = F[f * NN + n];
  }
}

// ------------------------------------------------- power iteration (rho(A))
// w[row[e]] += val[e] * v[col[e]]
__global__ void k_spmv(const int* __restrict__ row, const int* __restrict__ col,
                       const float* __restrict__ val, const float* __restrict__ v,
                       float* __restrict__ w) {
  int e = blockIdx.x * blockDim.x + threadIdx.x;
  if (e < NE) atomicAdd(&w[row[e]], val[e] * v[col[e]]);
}

__global__ void k_sumsq(const float* __restrict__ w, float* __restrict__ slot) {
  __shared__ float sm[256];
  int i = blockIdx.x * 256 + threadIdx.x;
  float x = (i < NN) ? w[i] : 0.0f;
  sm[threadIdx.x] = x * x;
  __syncthreads();
  for (int off = 128; off > 0; off >>= 1) {
    if ((int)threadIdx.x < off) sm[threadIdx.x] += sm[threadIdx.x + off];
    __syncthreads();
  }
  if (threadIdx.x == 0) atomicAdd(slot, sm[0]);
}

__global__ void k_normalize(const float* __restrict__ w, const float* __restrict__ slot,
                            float* __restrict__ v) {
  int i = blockIdx.x * blockDim.x + threadIdx.x;
  if (i < NN) v[i] = w[i] / (sqrtf(*slot) + 1e-12f);
}

// --------------------------------------------- row-wise L1-ball projection
template<int D>
__global__ void k_project(const float* __restrict__ W, const float* __restrict__ rho_sq,
                          float* __restrict__ Wp) {
  int r = threadIdx.x;
  if (r >= D) return;
  float kk = KAPPA_F / (sqrtf(*rho_sq) + 1e-12f);
  const float* wr = W + r * D;
  float a[128];
  float sum = 0.0f;
  for (int j = 0; j < D; ++j) { a[j] = fabsf(wr[j]); sum += a[j]; }
  if (sum <= kk) {
    for (int j = 0; j < D; ++j) Wp[r * D + j] = wr[j];
    return;
  }
  float sarr[128];                       // insertion sort, descending
  for (int j = 0; j < D; ++j) {
    float x = a[j];
    int m = j;
    while (m > 0 && sarr[m - 1] < x) { sarr[m] = sarr[m - 1]; --m; }
    sarr[m] = x;
  }
  float cs = 0.0f, cs_at = 0.0f;
  int nnz = 1;
  for (int j = 0; j < D; ++j) {
    cs += sarr[j];
    if (sarr[j] - (cs - kk) / (float)(j + 1) > 0.0f) { nnz = j + 1; cs_at = cs; }
  }
  float theta = fmaxf((cs_at - kk) / (float)nnz, 0.0f);
  for (int j = 0; j < D; ++j) {
    float wv = wr[j];
    float m = fmaxf(fabsf(wv) - theta, 0.0f);
    Wp[r * D + j] = (wv > 0.0f) ? m : ((wv < 0.0f) ? -m : 0.0f);
  }
}

// ------------------------------------------------- SpMM scatter: Y = A^T X
// Y[col[e], :] += val[e] * X[row[e], :]
template<int DOUT>
__global__ void k_spmm(const int* __restrict__ row, const int* __restrict__ col,
                       const float* __restrict__ val, const float* __restrict__ X,
                       float* __restrict__ Y) {
  constexpr int Q = DOUT / 4;
  int idx = blockIdx.x * blockDim.x + threadIdx.x;
  if (idx >= NE * Q) return;
  int e = idx / Q;
  int f = (idx - e * Q) * 4;
  int rs = row[e], cd = col[e];
  float vv = val[e];
  const float4 xv = *reinterpret_cast<const float4*>(X + (size_t)rs * DOUT + f);
  float* yp = Y + (size_t)cd * DOUT + f;
  atomicAdd(yp + 0, vv * xv.x);
  atomicAdd(yp + 1, vv * xv.y);
  atomicAdd(yp + 2, vv * xv.z);
  atomicAdd(yp + 3, vv * xv.w);
}

// ------------------------------------------------------- WMMA f32 GEMM
// C[NN, DOUT] = act( A[NN, K] @ Wt^T + (Add?) + (Bias?) ), Wt is [DOUT, K] row-major.
// One wave per 16-row tile, register-blocked over DOUT/16 n-tiles.
// ACT: 0=none 1=relu 2=elu
template<int K, int DOUT, int ACT, bool HAS_ADD, bool HAS_BIAS>
__launch_bounds__(128)
__global__ void k_gemm(const float* __restrict__ A, const float* __restrict__ Wt,
                       const float* __restrict__ Add, const float* __restrict__ Bias,
                       float* __restrict__ C) {
  constexpr int NT = DOUT / 16;
  const int lane = threadIdx.x & 31;
  const int tile = blockIdx.x * 4 + (threadIdx.x >> 5);
  if (tile >= MT) return;
  const int m0 = tile * 16;
  const int l15 = lane & 15;
  const int lh2 = (lane >> 4) * 2;     // K sub-offset: 0 (lanes 0-15) or 2 (lanes 16-31)

  v8f acc[NT];
#pragma unroll
  for (int t = 0; t < NT; ++t)
#pragma unroll
    for (int r = 0; r < 8; ++r) acc[t][r] = 0.0f;

  // A 16x4 f32 fragment: lane = M (mod 16), VGPR0 = K=lh2, VGPR1 = K=lh2+1
  const float* arow = A + (size_t)(m0 + l15) * K + lh2;
  for (int k0 = 0; k0 < K; k0 += 4) {
    v2f af;
    af.x = arow[k0];
    af.y = arow[k0 + 1];
#pragma unroll
    for (int t = 0; t < NT; ++t) {
      // B 4x16 f32 fragment: lane = N (mod 16); B[k][n] = Wt[n*K + k]
      const float* bp = Wt + (size_t)(t * 16 + l15) * K + k0 + lh2;
      v2f bf;
      bf.x = bp[0];
      bf.y = bp[1];
      acc[t] = __builtin_amdgcn_wmma_f32_16x16x4_f32(false, af, false, bf,
                                                     (short)0, acc[t], false, false);
    }
  }

  // C/D 16x16 f32: lanes 0-15 rows m0..m0+7, lanes 16-31 rows m0+8..m0+15
  const int mbase = m0 + ((lane >> 4) << 3);
#pragma unroll
  for (int r = 0; r < 8; ++r) {
    const int m = mbase + r;
#pragma unroll
    for (int t = 0; t < NT; ++t) {
      const int n = t * 16 + l15;
      float v = acc[t][r];
      if (HAS_ADD)  v += Add[(size_t)m * DOUT + n];
      if (HAS_BIAS) v += Bias[n];
      if (ACT == 1) v = fmaxf(v, 0.0f);
      else if (ACT == 2) v = (v > 0.0f) ? v : (__expf(v) - 1.0f);
      C[(size_t)m * DOUT + n] = v;
    }
  }
}

// ----------------------------------------------------------- one IGNN layer
template<int DIN_, int DOUT_, int FINAL_ACT>
static void run_layer(const int* row, const int* col, const float* val,
                      float* x, float* z, float* bt, float* s, float* wp,
                      const float* W, const float* Om, const float* Pw, const float* Pb,
                      const float* rho_sq, float* xnext, hipStream_t stream) {
  const int nzero = NN * DOUT_;
  const int zb = (nzero + 255) / 256;
  const int spmm_blocks = (NE * (DOUT_ / 4) + 255) / 256;

  // Wp = project_inf_norm(W, kappa/rho)
  k_project<DOUT_><<<1, DOUT_, 0, stream>>>(W, rho_sq, wp);
  // g = x @ Om^T   (into s)
  k_gemm<DIN_, DOUT_, 0, false, false><<<GEMM_BLOCKS, 128, 0, stream>>>(x, Om, nullptr, nullptr, s);
  // bt = A^T g
  k_zero<<<zb, 256, 0, stream>>>(bt, nzero);
  k_spmm<DOUT_><<<spmm_blocks, 256, 0, stream>>>(row, col, val, s, bt);
  // fixed point: Z = relu((A^T Z) Wp^T + bt), Z0 = 0
  k_zero<<<zb, 256, 0, stream>>>(z, nzero);
  for (int it = 0; it < 15; ++it) {
    k_zero<<<zb, 256, 0, stream>>>(s, nzero);
    k_spmm<DOUT_><<<spmm_blocks, 256, 0, stream>>>(row, col, val, z, s);
    k_gemm<DOUT_, DOUT_, 1, true, false><<<GEMM_BLOCKS, 128, 0, stream>>>(s, wp, bt, nullptr, z);
  }
  // xnext = act( Z + x @ Pw^T + Pb )
  k_gemm<DIN_, DOUT_, FINAL_ACT, true, true><<<GEMM_BLOCKS, 128, 0, stream>>>(x, Pw, z, Pb, xnext);
}

// ------------------------------------------------------------------- driver
extern "C" void kernel_launch(void* const* d_in, const int* in_sizes, int n_in,
                              void* d_out, int out_size, void* d_ws, size_t ws_size,
                              hipStream_t stream) {
  (void)in_sizes; (void)n_in; (void)out_size; (void)ws_size;
  const float* features = (const float*)d_in[0];
  const int*   row      = (const int*)d_in[1];
  const int*   col      = (const int*)d_in[2];
  const float* val      = (const float*)d_in[3];
  const float *W[5], *Om[5], *Pw[5], *Pb[5];
  for (int i = 0; i < 5; ++i) {
    W[i]  = (const float*)d_in[4 + 4 * i];
    Om[i] = (const float*)d_in[5 + 4 * i];
    Pw[i] = (const float*)d_in[6 + 4 * i];
    Pb[i] = (const float*)d_in[7 + 4 * i];
  }

  char* p = (char*)d_ws;
  const size_t big = (size_t)NN * 128 * sizeof(float);
  float* b0 = (float*)p; p += big;
  float* b1 = (float*)p; p += big;
  float* b2 = (float*)p; p += big;
  float* b3 = (float*)p; p += big;
  float* wp    = (float*)p; p += 128 * 128 * sizeof(float);
  float* pv    = (float*)p; p += ((size_t)NN + 64) * sizeof(float);
  float* pwv   = (float*)p; p += ((size_t)NN + 64) * sizeof(float);
  float* norms = (float*)p; p += 64 * sizeof(float);

  // ---- spectral radius: 50 power iterations + final norm (norms[50] = rho^2)
  k_zero<<<1, 64, 0, stream>>>(norms, 64);
  k_fill<<<(NN + 255) / 256, 256, 0, stream>>>(pv, NN, 1.0f / sqrtf((float)NN));
  for (int t = 0; t <= 50; ++t) {
    k_zero<<<(NN + 255) / 256, 256, 0, stream>>>(pwv, NN);
    k_spmv<<<(NE + 255) / 256, 256, 0, stream>>>(row, col, val, pv, pwv);
    k_sumsq<<<(NN + 255) / 256, 256, 0, stream>>>(pwv, norms + t);
    if (t < 50)
      k_normalize<<<(NN + 255) / 256, 256, 0, stream>>>(pwv, norms + t, pv);
  }
  const float* rho_sq = norms + 50;

  // ---- x = features^T (node-major)
  float* x = b0; float* z = b1; float* bt = b2; float* s = b3;
  k_transpose_feat<<<(NN * 128 + 255) / 256, 256, 0, stream>>>(features, x);

  // ---- 5 layers; xnext written into bt, then rotate x<->bt
  run_layer<128, 128, 2>(row, col, val, x, z, bt, s, wp, W[0], Om[0], Pw[0], Pb[0], rho_sq, bt, stream);
  { float* t_ = x; x = bt; bt = t_; }
  run_layer<128,  64, 2>(row, col, val, x, z, bt, s, wp, W[1], Om[1], Pw[1], Pb[1], rho_sq, bt, stream);
  { float* t_ = x; x = bt; bt = t_; }
  run_layer< 64,  64, 2>(row, col, val, x, z, bt, s, wp, W[2], Om[2], Pw[2], Pb[2], rho_sq, bt, stream);
  { float* t_ = x; x = bt; bt = t_; }
  run_layer< 64,  32, 2>(row, col, val, x, z, bt, s, wp, W[3], Om[3], Pw[3], Pb[3], rho_sq, bt, stream);
  { float* t_ = x; x = bt; bt = t_; }
  run_layer< 32,  16, 0>(row, col, val, x, z, bt, s, wp, W[4], Om[4], Pw[4], Pb[4], rho_sq,
                         (float*)d_out, stream);
}